// Com_class_mean_38439957300009
// MI455X (gfx1250) — compile-verified
//
#include <hip/hip_runtime.h>

// Class-mean replacement: segment-mean over (image, class, channel) + gather.
// Memory-bound: ~516 MB traffic -> ~22us floor @ 23.3 TB/s.
// Pass 1: one-hot GEMM via V_WMMA_F32_16X16X4_F32 (full f32 precision),
//         img staged global->LDS with async-to-LDS double buffering (ASYNCcnt).
// Pass 2: gather means, stream 256 MB of coalesced float4 stores.

typedef __attribute__((ext_vector_type(2))) float v2f;
typedef __attribute__((ext_vector_type(8))) float v8f;

#define CLASSES 21
#define BATCH   8
#define CH      128
#define HW      65536          // 256*256
#define EPSF    1e-8f

#define P1_CHUNK 2048          // pixels per workgroup in pass 1
#define P1_SUB   64            // pixels per LDS sub-tile
#define NSTAGE   (P1_CHUNK / P1_SUB)
#define PPAD     68            // 64 + 4 pad: 68 % 64 == 4 -> B reads bank-conflict-free

#define P3_CHUNK 1024          // pixels per workgroup in pass 3 (4 per thread)

#if __has_builtin(__builtin_amdgcn_global_load_async_to_lds_b128)
#define HAVE_ASYNC_LDS 1
typedef int v4i __attribute__((vector_size(16)));
typedef __attribute__((address_space(1))) v4i g_v4i;   // global (AS1) int4
typedef __attribute__((address_space(3))) v4i l_v4i;   // LDS (AS3) int4
#else
#define HAVE_ASYNC_LDS 0
#endif

template <int N>
__device__ __forceinline__ void async_wait_le() {
#if HAVE_ASYNC_LDS
#if __has_builtin(__builtin_amdgcn_s_wait_asynccnt)
    __builtin_amdgcn_s_wait_asynccnt(N);           // immediate required
#else
    asm volatile("s_wait_asynccnt %0" :: "i"(N) : "memory");
#endif
#endif
}

// ---------------------------------------------------------------- pass 0: zero ws
__global__ void zero_ws_kernel(float* ws, int n) {
    int i = blockIdx.x * blockDim.x + threadIdx.x;
    if (i < n) ws[i] = 0.0f;
}

// ---------------------------------------------------------------- pass 1: segment sums + counts
// grid = BATCH*32 WGs, 256 threads (8 waves). Wave w owns channels [16w, 16w+16).
// Each WG covers a 2048-pixel chunk of one image, double-buffered 64-pixel LDS tiles.
__global__ __launch_bounds__(256)
void seg_sum_wmma_kernel(const float* __restrict__ img, const int* __restrict__ gt,
                         float* __restrict__ sums, float* __restrict__ counts) {
    __shared__ float imgLds[2][CH * PPAD];  // [buf][channel][pixel], padded, 2x34 KB
    __shared__ int   gtLds[P1_CHUNK];
    __shared__ float cntLds[32];

    const int b     = blockIdx.x >> 5;            // image
    const int pbase = (blockIdx.x & 31) * P1_CHUNK;
    const int t     = threadIdx.x;
    const int lane  = t & 31;
    const int wave  = t >> 5;
    const int half  = lane >> 4;                  // K-group: lanes 16-31 use K+2
    const int m     = lane & 15;                  // A row (class) / B col (channel)
    const int c_l   = wave * 16 + m;              // this lane's channel for B/C/D

    const float* imgb = img + (size_t)b * CH * HW + pbase;
    const int cg = t >> 4;                        // 0..15: channel group for staging
    const int p4 = (t & 15) * 4;                  // 0..60: pixel quad for staging

    // Issue one stage's 8 async b128 copies (128ch x 64px) into buffer `buf`.
    auto issue_stage = [&](int stage, int buf) {
        #pragma unroll
        for (int i = 0; i < 8; ++i) {
            const int c = cg + 16 * i;
            const float* gsrc = imgb + (size_t)c * HW + stage * P1_SUB + p4;
            float* ldst = &imgLds[buf][c * PPAD + p4];
#if HAVE_ASYNC_LDS
            __builtin_amdgcn_global_load_async_to_lds_b128(
                (g_v4i*)gsrc, (l_v4i*)ldst, 0, 0);
#else
            *(float4*)ldst = *(const float4*)gsrc;   // sync fallback
#endif
        }
    };

    // ---- gt chunk -> LDS, plus per-WG class histogram
    if (t < 32) cntLds[t] = 0.0f;
    issue_stage(0, 0);                            // prefetch stage 0 while doing gt work
    __syncthreads();
    {
        const int* gtp = gt + (size_t)b * HW + pbase;
        int labs[8];
        #pragma unroll
        for (int i = 0; i < 8; ++i) {
            labs[i] = gtp[t + 256 * i];           // coalesced
            gtLds[t + 256 * i] = labs[i];
        }
        #pragma unroll
        for (int i = 0; i < 8; ++i) atomicAdd(&cntLds[labs[i]], 1.0f);
    }

    v8f acc0 = {};   // classes 0..15  x channels [16w,16w+16)
    v8f acc1 = {};   // classes 16..31 (only 16..20 real; rest get zero one-hot)

    for (int stage = 0; stage < NSTAGE; ++stage) {
        const int buf = stage & 1;
        if (stage + 1 < NSTAGE) {
            // Buffer buf^1 was last consumed at stage-1; the barrier at the end of
            // that iteration makes this overwrite safe. ASYNCcnt completes in order,
            // so waiting <=8 guarantees THIS stage's 8 copies have landed.
            issue_stage(stage + 1, buf ^ 1);
            async_wait_le<8>();
        } else {
            async_wait_le<0>();
        }
        __syncthreads();                          // all waves' tiles visible

        const float* tile = imgLds[buf];
        const int gtoff = stage * P1_SUB;
        #pragma unroll 4
        for (int p4i = 0; p4i < P1_SUB / 4; ++p4i) {
            const int pl = 4 * p4i + 2 * half;    // K pixel pair for this lane
            const int lab0 = gtLds[gtoff + pl];       // LDS broadcast within half-wave
            const int lab1 = gtLds[gtoff + pl + 1];

            v2f bv;                                // B: img[pixel K][channel N]
            bv.x = tile[c_l * PPAD + pl];          // banks 4c+p: conflict-free, PPAD=68
            bv.y = tile[c_l * PPAD + pl + 1];

            v2f a0, a1;                            // A: one-hot[class M][pixel K]
            a0.x = (lab0 == m)      ? 1.0f : 0.0f;
            a0.y = (lab1 == m)      ? 1.0f : 0.0f;
            a1.x = (lab0 == m + 16) ? 1.0f : 0.0f;
            a1.y = (lab1 == m + 16) ? 1.0f : 0.0f;

            acc0 = __builtin_amdgcn_wmma_f32_16x16x4_f32(false, a0, false, bv,
                                                         (short)0, acc0, false, false);
            acc1 = __builtin_amdgcn_wmma_f32_16x16x4_f32(false, a1, false, bv,
                                                         (short)0, acc1, false, false);
        }
        __syncthreads();                          // done with this buffer
    }

    // ---- merge per-wave 16x16 accumulators into global sums[b][cls][c]
    float* sb = sums + (size_t)b * CLASSES * CH;
    #pragma unroll
    for (int v = 0; v < 8; ++v) {
        const int cls0 = v + 8 * half;                         // 0..15, always valid
        atomicAdd(&sb[cls0 * CH + c_l], acc0[v]);
        const int cls1 = 16 + v + 8 * half;                    // valid only 16..20
        if (cls1 < CLASSES) atomicAdd(&sb[cls1 * CH + c_l], acc1[v]);
    }
    __syncthreads();
    if (t < CLASSES) atomicAdd(&counts[b * CLASSES + t], cntLds[t]);
}

// ---------------------------------------------------------------- pass 2: gather means -> out
// grid = BATCH*64 WGs, 256 threads, 4 consecutive pixels per thread, loop over 128 channels.
__global__ __launch_bounds__(256)
void gather_out_kernel(const float* __restrict__ sums, const float* __restrict__ counts,
                       const int* __restrict__ gt, float* __restrict__ out) {
    __shared__ float mLds[CH * 32];   // means, layout [c][32]: gather banks = cls + 32*(c&1)

    const int b     = blockIdx.x >> 6;
    const int pbase = (blockIdx.x & 63) * P3_CHUNK;
    const int t     = threadIdx.x;

    for (int idx = t; idx < CH * 32; idx += 256) {
        const int c = idx >> 5, cls = idx & 31;
        float v = 0.0f;
        if (cls < CLASSES)
            v = sums[((size_t)b * CLASSES + cls) * CH + c] /
                (counts[b * CLASSES + cls] + EPSF);
        mLds[idx] = v;
    }
    __syncthreads();

    const int4 lab = *(const int4*)(gt + (size_t)b * HW + pbase + 4 * t);   // 16B aligned
    float* ob = out + (size_t)b * CH * HW + pbase + 4 * t;

    #pragma unroll 4
    for (int c = 0; c < CH; ++c) {
        const float* mrow = &mLds[c * 32];
        float4 o;
        o.x = mrow[lab.x]; o.y = mrow[lab.y]; o.z = mrow[lab.z]; o.w = mrow[lab.w];
        *(float4*)(ob + (size_t)c * HW) = o;     // coalesced 16B stores
    }
}

// ---------------------------------------------------------------- launch
extern "C" void kernel_launch(void* const* d_in, const int* in_sizes, int n_in,
                              void* d_out, int out_size, void* d_ws, size_t ws_size,
                              hipStream_t stream) {
    const float* img = (const float*)d_in[0];   // [8,128,256,256] f32
    const int*   gt  = (const int*)d_in[1];     // [8,1,256,256] i32
    float*       out = (float*)d_out;           // [8,128,256,256] f32

    float* sums   = (float*)d_ws;                         // [8][21][128]
    float* counts = sums + BATCH * CLASSES * CH;          // [8][21]
    const int nz = BATCH * CLASSES * CH + BATCH * CLASSES;

    zero_ws_kernel<<<(nz + 255) / 256, 256, 0, stream>>>((float*)d_ws, nz);
    seg_sum_wmma_kernel<<<BATCH * 32, 256, 0, stream>>>(img, gt, sums, counts);
    gather_out_kernel<<<BATCH * 64, 256, 0, stream>>>(sums, counts, gt, out);
}